// GNNClassifier_69380901699675
// MI455X (gfx1250) — compile-verified
//
#include <hip/hip_runtime.h>
#include <math.h>

// Problem constants (from reference)
constexpr int NGRAPH = 1000;
constexpr int NP     = 100;    // nodes per graph
constexpr int NEPG   = 1200;   // edges per graph
constexpr int FIN    = 16;
constexpr int FHID   = 64;
constexpr int FOUT   = 5;

constexpr int NTHREADS = 256;
constexpr int NWAVES   = 8;

// LDS geometry
constexpr int MPAD = 112;              // 7 * 16 row tiles (rows 100..111 stay zero)
constexpr int XSTR = 3 * FHID + 4;     // 196 dwords row stride (mod 64 banks = 4 -> conflict free)
constexpr int PSTR = 160;              // pair-packed W row stride in dwords (pair step = 160 ≡ 32 mod 64
                                       //  -> the two half-waves use disjoint 32-bank sets; b64 conflict-free)

constexpr int XT_FLOATS = MPAD * XSTR;          // 21952
constexpr int WP_FLOATS = (192 / 2) * PSTR;     // 96 K-pairs * 160 = 15360
constexpr int DINV_OFF  = XT_FLOATS + WP_FLOATS;
constexpr int BVEC_OFF  = DINV_OFF + MPAD;
constexpr int GVEC_OFF  = BVEC_OFF + FHID;
constexpr int EDGE_OFF  = GVEC_OFF + FHID;      // ints from here
constexpr int SMEM_FLOATS = EDGE_OFF + 2 * NEPG;
constexpr size_t SMEM_BYTES = (size_t)SMEM_FLOATS * 4; // 159,808 B (<=160KB: 2 WGs / 320KB WGP)

typedef float v2f __attribute__((ext_vector_type(2)));
typedef float v8f __attribute__((ext_vector_type(8)));

// ---- Chebyshev expansion: Xt[:,0:F]=X0 given; build X1=-AX0 into [F,2F), X2=-2AX1-X0 into [2F,3F)
template <int F>
__device__ __forceinline__ void cheb_expand(float* Xt, const float* dinv,
                                            const int* es, const int* ed, int tid) {
  // zero X1/X2 regions (rows < NP only; padded rows stay zero forever)
  for (int i = tid; i < NP * 2 * F; i += NTHREADS) {
    int r = i / (2 * F), c = i % (2 * F);
    Xt[r * XSTR + F + c] = 0.0f;
  }
  __syncthreads();

  constexpr int C4 = F / 4;
  // scatter: agg1[d] += X0[s] * dinv[s]
  for (int w = tid; w < NEPG * C4; w += NTHREADS) {
    int e = w / C4;
    int c = (w % C4) * 4;
    int s = es[e], d = ed[e];
    float ds = dinv[s];
    const float* xs = &Xt[s * XSTR + c];
    float* xd = &Xt[d * XSTR + F + c];
    atomicAdd(&xd[0], xs[0] * ds);
    atomicAdd(&xd[1], xs[1] * ds);
    atomicAdd(&xd[2], xs[2] * ds);
    atomicAdd(&xd[3], xs[3] * ds);
  }
  __syncthreads();
  // X1 = -dinv * agg1
  for (int i = tid; i < NP * F; i += NTHREADS) {
    int r = i / F, c = i % F;
    Xt[r * XSTR + F + c] *= -dinv[r];
  }
  __syncthreads();
  // scatter: agg2[d] += X1[s] * dinv[s]
  for (int w = tid; w < NEPG * C4; w += NTHREADS) {
    int e = w / C4;
    int c = (w % C4) * 4;
    int s = es[e], d = ed[e];
    float ds = dinv[s];
    const float* xs = &Xt[s * XSTR + F + c];
    float* xd = &Xt[d * XSTR + 2 * F + c];
    atomicAdd(&xd[0], xs[0] * ds);
    atomicAdd(&xd[1], xs[1] * ds);
    atomicAdd(&xd[2], xs[2] * ds);
    atomicAdd(&xd[3], xs[3] * ds);
  }
  __syncthreads();
  // X2 = -2*dinv*agg2 - X0
  for (int i = tid; i < NP * F; i += NTHREADS) {
    int r = i / F, c = i % F;
    Xt[r * XSTR + 2 * F + c] = -2.0f * dinv[r] * Xt[r * XSTR + 2 * F + c] - Xt[r * XSTR + c];
  }
  __syncthreads();
}

// ---- GEMM: (MPADxKDIM Xt) @ (KDIMx64 W) + b, ReLU, written back into Xt cols [0,64)
// W staged pair-packed: element (k,n) -> Wl[(k>>1)*PSTR + 2n + (k&1)], so a lane's B fragment
// {W[k][n], W[k+1][n]} is one aligned 8-byte LDS load.
template <int KDIM>
__device__ __forceinline__ void gemm_relu(float* Xt, float* Wl, float* bvec,
                                          const float* __restrict__ W,
                                          const float* __restrict__ b, int tid) {
  for (int i = tid; i < KDIM * FHID; i += NTHREADS) {
    int k = i / FHID, n = i % FHID;
    Wl[(k >> 1) * PSTR + 2 * n + (k & 1)] = W[i];
  }
  for (int i = tid; i < FHID; i += NTHREADS) bvec[i] = b[i];
  __syncthreads();

  const int wave = tid >> 5;
  const int lane = tid & 31;
  const int half = lane >> 4;   // 0: K={k0,k0+1}, 1: K={k0+2,k0+3}
  const int l    = lane & 15;

  constexpr int NTILES = 7 * 4; // M tiles x N tiles
  constexpr int PV = PSTR / 2;  // pair-row stride in v2f units
  v8f acc[4] = {};

#pragma unroll
  for (int t = 0; t < 4; ++t) {
    int tile = wave + t * NWAVES;
    if (tile < NTILES) {
      int m0 = (tile >> 2) * 16;
      int n0 = (tile & 3) * 16;
      // A[M=m0+l][k0+2h], [k0+2h+1] : one b64 per k-step (row base even, k0+2h even)
      const v2f* ap = (const v2f*)&Xt[(m0 + l) * XSTR];
      // B pair row (k0/2+h), column n0+l : one b64 per k-step
      const v2f* bp = ((const v2f*)Wl) + (n0 + l);
      v8f c = {};
#pragma unroll 4
      for (int kk = 0; kk < KDIM / 2; kk += 2) {   // kk = k0/2
        v2f a  = ap[kk + half];
        v2f bf = bp[(kk + half) * PV];
        c = __builtin_amdgcn_wmma_f32_16x16x4_f32(false, a, false, bf,
                                                  (short)0, c, false, false);
      }
      acc[t] = c;
    }
  }
  __syncthreads();  // all Xt reads done before in-place overwrite of cols [0,64)

#pragma unroll
  for (int t = 0; t < 4; ++t) {
    int tile = wave + t * NWAVES;
    if (tile < NTILES) {
      int m0 = (tile >> 2) * 16;
      int n0 = (tile & 3) * 16;
      float bb = bvec[n0 + l];
#pragma unroll
      for (int r = 0; r < 8; ++r) {
        int row = m0 + r + 8 * half;   // D layout: VGPR r -> M=r (lanes 0-15), M=r+8 (lanes 16-31)
        if (row < NP) {
          float v = acc[t][r] + bb;
          Xt[row * XSTR + n0 + l] = v > 0.0f ? v : 0.0f;
        }
      }
    }
  }
  __syncthreads();
}

__global__ __launch_bounds__(NTHREADS)
void gnn_cheb_kernel(const float* __restrict__ feat,
                     const int* __restrict__ src, const int* __restrict__ dst,
                     const float* __restrict__ W1, const float* __restrict__ b1,
                     const float* __restrict__ W2, const float* __restrict__ b2,
                     const float* __restrict__ W3, const float* __restrict__ b3,
                     const float* __restrict__ Wfc, const float* __restrict__ bfc,
                     float* __restrict__ out) {
  extern __shared__ float smem[];
  float* Xt   = smem;
  float* Wl   = smem + XT_FLOATS;
  float* dinv = smem + DINV_OFF;
  float* bvec = smem + BVEC_OFF;
  float* gvec = smem + GVEC_OFF;
  int*   es   = (int*)(smem + EDGE_OFF);
  int*   ed   = es + NEPG;

  const int g = blockIdx.x;
  const int tid = threadIdx.x;
  const int nodeBase = g * NP;
  const int edgeBase = g * NEPG;

  // warm L2 with the big (shared) weight matrices while we do per-graph setup
  for (int i = tid; i < (3 * FHID * FHID) / 32; i += NTHREADS) {
    __builtin_prefetch(&W2[i * 32], 0, 1);
    __builtin_prefetch(&W3[i * 32], 0, 1);
  }

  // zero Xt (incl. padded rows 100..111) and degree accumulator
  for (int i = tid; i < XT_FLOATS; i += NTHREADS) Xt[i] = 0.0f;
  for (int i = tid; i < MPAD; i += NTHREADS) dinv[i] = 0.0f;
  __syncthreads();

  // load edges (local ids), accumulate in-degree; load features into Xt cols [0,16)
  for (int e = tid; e < NEPG; e += NTHREADS) {
    int s = src[edgeBase + e] - nodeBase;
    int d = dst[edgeBase + e] - nodeBase;
    es[e] = s;
    ed[e] = d;
    atomicAdd(&dinv[d], 1.0f);
  }
  for (int i = tid; i < NP * FIN; i += NTHREADS) {
    int r = i >> 4, c = i & 15;
    Xt[r * XSTR + c] = feat[(nodeBase + r) * FIN + c];
  }
  __syncthreads();
  for (int i = tid; i < NP; i += NTHREADS) {
    float dd = dinv[i];
    dd = dd < 1.0f ? 1.0f : dd;
    dinv[i] = 1.0f / sqrtf(dd);
  }
  __syncthreads();

  // layer 1: F=16, K-dim 48
  cheb_expand<FIN>(Xt, dinv, es, ed, tid);
  gemm_relu<3 * FIN>(Xt, Wl, bvec, W1, b1, tid);
  // layer 2: F=64, K-dim 192
  cheb_expand<FHID>(Xt, dinv, es, ed, tid);
  gemm_relu<3 * FHID>(Xt, Wl, bvec, W2, b2, tid);
  // layer 3
  cheb_expand<FHID>(Xt, dinv, es, ed, tid);
  gemm_relu<3 * FHID>(Xt, Wl, bvec, W3, b3, tid);

  // mean pool over the graph's NP nodes
  for (int f = tid; f < FHID; f += NTHREADS) {
    float s = 0.0f;
    for (int i = 0; i < NP; ++i) s += Xt[i * XSTR + f];
    gvec[f] = s * (1.0f / (float)NP);
  }
  __syncthreads();

  // final FC: (64) @ (64x5) + bfc
  if (tid < FOUT) {
    float s = bfc[tid];
    for (int f = 0; f < FHID; ++f) s += gvec[f] * Wfc[f * FOUT + tid];
    out[g * FOUT + tid] = s;
  }
}

extern "C" void kernel_launch(void* const* d_in, const int* in_sizes, int n_in,
                              void* d_out, int out_size, void* d_ws, size_t ws_size,
                              hipStream_t stream) {
  const float* feat = (const float*)d_in[0];
  const int*   src  = (const int*)d_in[1];
  const int*   dst  = (const int*)d_in[2];
  // d_in[3] = graph_ids (implied by block structure), d_in[4] = n_graphs (scalar)
  const float* W1  = (const float*)d_in[5];
  const float* b1  = (const float*)d_in[6];
  const float* W2  = (const float*)d_in[7];
  const float* b2  = (const float*)d_in[8];
  const float* W3  = (const float*)d_in[9];
  const float* b3  = (const float*)d_in[10];
  const float* Wfc = (const float*)d_in[11];
  const float* bfc = (const float*)d_in[12];
  float* out = (float*)d_out;

  int nGraphs = in_sizes[1] / NEPG;  // = 1000
  hipLaunchKernelGGL(gnn_cheb_kernel, dim3(nGraphs), dim3(NTHREADS), SMEM_BYTES, stream,
                     feat, src, dst, W1, b1, W2, b2, W3, b3, Wfc, bfc, out);
}